// SemanticRND_26860725469335
// MI455X (gfx1250) — compile-verified
//
#include <hip/hip_runtime.h>
#include <stdint.h>

// Stratified min-max normalization, group_size = 16 (fixed by the reference).
// Memory-bound: 288 MB traffic -> ~12.4 us at 23.3 TB/s. One lane per group,
// all-b128 non-temporal traffic, no LDS/barriers, wave32-dense.

typedef float    v4f __attribute__((ext_vector_type(4)));
typedef uint32_t v4u __attribute__((ext_vector_type(4)));

#define CURIOSITY_COEF 0.1f
#define EPS_F 1e-8f

__global__ __launch_bounds__(256)
void stratnorm16_kernel(const float* __restrict__ rewards,
                        const unsigned char* __restrict__ corr,
                        float* __restrict__ out,
                        int nGroups) {
    long g = (long)blockIdx.x * blockDim.x + threadIdx.x;
    if (g >= nGroups) return;

    const long base = g * 16;

    // Warm L2 one block-tile ahead (global_prefetch_b8; speculative, OOB-safe).
    __builtin_prefetch(rewards + base + 256 * 16, 0, 1);

    const v4f* rp = (const v4f*)(rewards + base);
    const v4u* mp = (const v4u*)(corr + base);

    // 4x global_load_b128 (rewards) + 1x global_load_b128 (16 mask bytes), TH=NT.
    v4f rv0 = __builtin_nontemporal_load(rp + 0);
    v4f rv1 = __builtin_nontemporal_load(rp + 1);
    v4f rv2 = __builtin_nontemporal_load(rp + 2);
    v4f rv3 = __builtin_nontemporal_load(rp + 3);
    v4u mw  = __builtin_nontemporal_load(mp);

    float r[16] = { rv0.x, rv0.y, rv0.z, rv0.w,
                    rv1.x, rv1.y, rv1.z, rv1.w,
                    rv2.x, rv2.y, rv2.z, rv2.w,
                    rv3.x, rv3.y, rv3.z, rv3.w };
    uint32_t mword[4] = { mw.x, mw.y, mw.z, mw.w };

    // Per-stratum min/max/count (private register math, fully unrolled).
    float minc =  __builtin_inff(), maxc = -__builtin_inff();
    float mini =  __builtin_inff(), maxi = -__builtin_inff();
    int cc = 0;
    #pragma unroll
    for (int k = 0; k < 16; ++k) {
        bool m = ((mword[k >> 2] >> ((k & 3) * 8)) & 0xFFu) != 0u;
        float x = r[k];
        if (m) {
            cc += 1;
            minc = fminf(minc, x);
            maxc = fmaxf(maxc, x);
        } else {
            mini = fminf(mini, x);
            maxi = fmaxf(maxi, x);
        }
    }
    int ci = 16 - cc;

    // One exact IEEE division per stratum (empty stratum -> garbage scale,
    // but no element selects it, matching the reference's discard semantics).
    float scc = 1.0f / (maxc - minc + EPS_F);
    float sci = 1.0f / (maxi - mini + EPS_F);

    v4f o[4];
    float* of = (float*)o;
    #pragma unroll
    for (int k = 0; k < 16; ++k) {
        bool m = ((mword[k >> 2] >> ((k & 3) * 8)) & 0xFFu) != 0u;
        float x = r[k];
        float n_c = (cc == 1) ? 0.5f : (x - minc) * scc;
        float n_i = (ci == 1) ? 0.5f : (x - mini) * sci;
        float n   = m ? n_c : n_i;
        of[k] = m ? (CURIOSITY_COEF * n) : (-CURIOSITY_COEF * n);
    }

    // 4x global_store_b128, TH=NT (single-use output, keep L2 clean).
    v4f* op = (v4f*)(out + base);
    __builtin_nontemporal_store(o[0], op + 0);
    __builtin_nontemporal_store(o[1], op + 1);
    __builtin_nontemporal_store(o[2], op + 2);
    __builtin_nontemporal_store(o[3], op + 3);
}

extern "C" void kernel_launch(void* const* d_in, const int* in_sizes, int n_in,
                              void* d_out, int out_size, void* d_ws, size_t ws_size,
                              hipStream_t stream) {
    const float*         rewards = (const float*)d_in[0];
    const unsigned char* corr    = (const unsigned char*)d_in[1];  // jax bool = 1 byte
    float*               out     = (float*)d_out;

    const int n       = in_sizes[0];
    const int nGroups = n / 16;          // group_size fixed at 16 by the reference

    const int block = 256;               // 8 wave32 waves per workgroup
    const int grid  = (nGroups + block - 1) / block;

    stratnorm16_kernel<<<grid, block, 0, stream>>>(rewards, corr, out, nGroups);
}